// MATEHead_20005957665589
// MI455X (gfx1250) — compile-verified
//
#include <hip/hip_runtime.h>
#include <hip/hip_bf16.h>

// ---------------------------------------------------------------------------
// MATEHead: emissions = relu(seq @ fc_w^T + fc_b) @ cls_w^T + cls_b -> CRF NLL.
// GEMM1 (34.4 GFLOP fp32) via V_WMMA_F32_16X16X4_F32; GEMM2 (L=3) fused into
// the epilogue.  Double-buffered LDS staging fed by
// GLOBAL_LOAD_ASYNC_TO_LDS_B128 (ASYNCcnt) so the DMA of chunk c+1 overlaps
// the 32 WMMAs of chunk c; only s_wait_asynccnt + one barrier per chunk.
// ---------------------------------------------------------------------------

typedef __attribute__((ext_vector_type(2))) float v2f;
typedef __attribute__((ext_vector_type(4))) float v4f;
typedef __attribute__((ext_vector_type(8))) float v8f;
typedef __attribute__((ext_vector_type(4))) int   v4i;

#define BSZ 32
#define SSZ 512
#define DSZ 1024
#define HSZ 1024
#define LSZ 3
#define MSZ (BSZ * SSZ)          // 16384 rows

#define MB  128                  // rows per workgroup
#define HB  128                  // H columns per hb-iteration
#define KC  32                   // K chunk staged in LDS
#define TPB 512                  // 16 waves (wave32)
#define STR (KC + 4)             // 36 floats: 144B rows -> 16B aligned, banks conflict-free
#define DBUF (2 * MB * STR)      // floats per double-buffer slot (A tile + B tile)
#define NCH ((HSZ / 2 / HB) * (DSZ / KC))   // 4 * 32 = 128 chunks per WG

#if __has_builtin(__builtin_amdgcn_global_load_async_to_lds_b128)
#define HAVE_ASYNC 1
#else
#define HAVE_ASYNC 0
#endif

__device__ __forceinline__ void copy_b128_to_lds(const float* g, float* l) {
#if HAVE_ASYNC
  // signature (from hipcc diagnostic): (v4i as(1)*, v4i as(3)*, imm offset, imm cpol)
  __builtin_amdgcn_global_load_async_to_lds_b128(
      (__attribute__((address_space(1))) v4i*)g,
      (__attribute__((address_space(3))) v4i*)l,
      0, 0);
#else
  *(v4f*)l = *(const v4f*)g;     // synchronous fallback (still double-buffered)
#endif
}

__device__ __forceinline__ void wait_async_done() {
#if HAVE_ASYNC
#if __has_builtin(__builtin_amdgcn_s_wait_asynccnt)
  __builtin_amdgcn_s_wait_asynccnt(0);
#else
  asm volatile("s_wait_asynccnt 0x0" ::: "memory");
#endif
#endif
}

// ---------------------------------------------------------------------------
// emissions workspace init: emis[m][l] = cls_b[l]
// ---------------------------------------------------------------------------
__global__ void init_emis_kernel(float* __restrict__ emis,
                                 const float* __restrict__ cls_b) {
  int i = blockIdx.x * blockDim.x + threadIdx.x;
  if (i < MSZ * LSZ) emis[i] = cls_b[i % LSZ];
}

// ---------------------------------------------------------------------------
// Fused GEMM1 + relu + GEMM2.  grid = (MSZ/MB, 2): x = 128-row block,
// y = half of H.  Each wave: 2x2 grid of 16x16 WMMA tiles.
// ---------------------------------------------------------------------------
__global__ __launch_bounds__(TPB)
void fused_mlp_kernel(const float* __restrict__ seq,   // [MSZ][DSZ]
                      const float* __restrict__ fcw,   // [HSZ][DSZ]
                      const float* __restrict__ fcb,   // [HSZ]
                      const float* __restrict__ clsw,  // [LSZ][HSZ]
                      float* __restrict__ emis)        // [MSZ][LSZ]
{
  __shared__ float smem[2 * DBUF];                     // 73,728 B of 320 KB WGP LDS

  const int tid  = threadIdx.x;
  const int lane = tid & 31;
  const int wv   = tid >> 5;        // wave 0..15
  const int ln   = lane & 15;       // N / M-row selector within fragment
  const int hi   = lane >> 4;       // K-pair (A/B frags) / +8 M rows (C)
  const int rt0  = (wv >> 2) * 2;   // row tiles {0,1},{2,3},{4,5},{6,7}
  const int ct0  = (wv & 3) * 2;    // col tiles likewise
  const int m0   = blockIdx.x * MB;
  const int hbase = blockIdx.y * (HSZ / 2);

  // fragment offsets within a buffer (floats)
  const int aoff0 = (rt0 * 16 + ln) * STR + 2 * hi;
  const int aoff1 = ((rt0 + 1) * 16 + ln) * STR + 2 * hi;
  const int boff0 = MB * STR + (ct0 * 16 + ln) * STR + 2 * hi;
  const int boff1 = MB * STR + ((ct0 + 1) * 16 + ln) * STR + 2 * hi;

  // issue async DMA for chunk c into buffer slot `which`
  auto issueChunk = [&](int c, int which) {
    const int hb = c >> 5;
    const int kc = (c & 31) * KC;
    const int h0 = hbase + hb * HB;
    float* bufA = smem + which * DBUF;
    float* bufB = bufA + MB * STR;
    #pragma unroll
    for (int j = 0; j < 2; ++j) {
      int idx = tid + j * TPB;            // 0..1023
      int row = idx >> 3;                 // 0..127
      int c4  = (idx & 7) * 4;            // 0,4,...,28 (16B granules)
      copy_b128_to_lds(&seq[(size_t)(m0 + row) * DSZ + kc + c4],
                       &bufA[row * STR + c4]);
      copy_b128_to_lds(&fcw[(size_t)(h0 + row) * DSZ + kc + c4],
                       &bufB[row * STR + c4]);
    }
  };

  float em[LSZ][2][8] = {};               // per-lane emission partials

  v8f acc[2][2];
  acc[0][0] = (v8f){}; acc[0][1] = (v8f){};
  acc[1][0] = (v8f){}; acc[1][1] = (v8f){};

  issueChunk(0, 0);

  for (int c = 0; c < NCH; ++c) {
    const int cur = c & 1;
    wait_async_done();                    // chunk c landed (this wave's DMAs)
    __syncthreads();                      // all waves: chunk c ready, c-1 reads done
    if (c + 1 < NCH) issueChunk(c + 1, cur ^ 1);   // overlaps compute below

    const float* base = smem + cur * DBUF;
    #pragma unroll
    for (int k = 0; k < KC; k += 4) {
      v2f a0 = *(const v2f*)(base + aoff0 + k);     // ds_load_b64, conflict-free
      v2f a1 = *(const v2f*)(base + aoff1 + k);
      v2f b0 = *(const v2f*)(base + boff0 + k);
      v2f b1 = *(const v2f*)(base + boff1 + k);
      acc[0][0] = __builtin_amdgcn_wmma_f32_16x16x4_f32(
          false, a0, false, b0, (short)0, acc[0][0], false, false);
      acc[0][1] = __builtin_amdgcn_wmma_f32_16x16x4_f32(
          false, a0, false, b1, (short)0, acc[0][1], false, false);
      acc[1][0] = __builtin_amdgcn_wmma_f32_16x16x4_f32(
          false, a1, false, b0, (short)0, acc[1][0], false, false);
      acc[1][1] = __builtin_amdgcn_wmma_f32_16x16x4_f32(
          false, a1, false, b1, (short)0, acc[1][1], false, false);
    }

    if ((c & 31) == 31) {
      // hb finished: bias + relu + contract with cls_w (fused GEMM2).
      // C layout: element r of acc <-> M = r + 8*hi, N = ln.
      const int h0 = hbase + (c >> 5) * HB;
      #pragma unroll
      for (int j = 0; j < 2; ++j) {
        int n = h0 + (ct0 + j) * 16 + ln;
        float bias = fcb[n];
        float w0 = clsw[0 * HSZ + n];
        float w1 = clsw[1 * HSZ + n];
        float w2 = clsw[2 * HSZ + n];
        #pragma unroll
        for (int i = 0; i < 2; ++i) {
          #pragma unroll
          for (int r = 0; r < 8; ++r) {
            float v = acc[i][j][r] + bias;
            v = v > 0.0f ? v : 0.0f;               // relu
            em[0][i][r] += v * w0;
            em[1][i][r] += v * w1;
            em[2][i][r] += v * w2;
          }
        }
      }
      acc[0][0] = (v8f){}; acc[0][1] = (v8f){};
      acc[1][0] = (v8f){}; acc[1][1] = (v8f){};
    }
  }

  // Reduce emission partials: 16 N-lanes + 4 col-tile waves via LDS atomics,
  // then the 2 H-halves via global f32 atomics.
  __syncthreads();
  float* emS = smem;                                 // reuse staging space
  for (int idx = tid; idx < MB * LSZ; idx += TPB) emS[idx] = 0.0f;
  __syncthreads();
  #pragma unroll
  for (int i = 0; i < 2; ++i) {
    #pragma unroll
    for (int r = 0; r < 8; ++r) {
      int lrow = (rt0 + i) * 16 + r + 8 * hi;
      atomicAdd(&emS[lrow * LSZ + 0], em[0][i][r]);
      atomicAdd(&emS[lrow * LSZ + 1], em[1][i][r]);
      atomicAdd(&emS[lrow * LSZ + 2], em[2][i][r]);
    }
  }
  __syncthreads();
  for (int idx = tid; idx < MB * LSZ; idx += TPB)
    atomicAdd(&emis[(size_t)m0 * LSZ + idx], emS[idx]);
}

// ---------------------------------------------------------------------------
// CRF mean-NLL: one wave, lane b = batch b; serial over S with L=3 states.
// ---------------------------------------------------------------------------
__device__ __forceinline__ float lse3(float x, float y, float z) {
  float m = fmaxf(x, fmaxf(y, z));
  return m + logf(expf(x - m) + expf(y - m) + expf(z - m));
}

__global__ void crf_nll_kernel(const float* __restrict__ emis,   // [B][S][L]
                               const int* __restrict__ tags,     // [B][S]
                               const unsigned char* __restrict__ mask, // [B][S]
                               const float* __restrict__ st,     // [L]
                               const float* __restrict__ tr,     // [L][L]
                               const float* __restrict__ et,     // [L]
                               float* __restrict__ out)
{
  const int b = threadIdx.x;                       // 0..31
  const float* em = emis + (size_t)b * SSZ * LSZ;
  const int* tg = tags + (size_t)b * SSZ;
  const unsigned char* mk = mask + (size_t)b * SSZ;

  float t00 = tr[0], t01 = tr[1], t02 = tr[2];
  float t10 = tr[3], t11 = tr[4], t12 = tr[5];
  float t20 = tr[6], t21 = tr[7], t22 = tr[8];

  int prev = tg[0];                                // mask[:,0] forced True
  float score = st[prev] + em[prev];
  float a0 = st[0] + em[0];
  float a1 = st[1] + em[1];
  float a2 = st[2] + em[2];
  int lastTag = prev;

  for (int t = 1; t < SSZ; ++t) {
    const float* e = em + t * LSZ;
    float e0 = e[0], e1 = e[1], e2 = e[2];
    int cur = tg[t];
    bool on = mk[t] != 0;
    if (on) {
      score += tr[prev * LSZ + cur] + e[cur];
      lastTag = cur;
    }
    prev = cur;                                    // tag pairs advance regardless
    float n0 = lse3(a0 + t00, a1 + t10, a2 + t20) + e0;
    float n1 = lse3(a0 + t01, a1 + t11, a2 + t21) + e1;
    float n2 = lse3(a0 + t02, a1 + t12, a2 + t22) + e2;
    if (on) { a0 = n0; a1 = n1; a2 = n2; }
  }
  score += et[lastTag];
  float part = lse3(a0 + et[0], a1 + et[1], a2 + et[2]);
  float llh = score - part;

  for (int off = 16; off > 0; off >>= 1) llh += __shfl_xor(llh, off);
  if (b == 0) out[0] = -(llh / (float)BSZ);
}

// ---------------------------------------------------------------------------
extern "C" void kernel_launch(void* const* d_in, const int* in_sizes, int n_in,
                              void* d_out, int out_size, void* d_ws, size_t ws_size,
                              hipStream_t stream) {
  const float* seq          = (const float*)d_in[0];
  const int* labels         = (const int*)d_in[1];
  const unsigned char* mask = (const unsigned char*)d_in[2];
  const float* fcw          = (const float*)d_in[3];
  const float* fcb          = (const float*)d_in[4];
  const float* clsw         = (const float*)d_in[5];
  const float* clsb         = (const float*)d_in[6];
  const float* st           = (const float*)d_in[7];
  const float* tr           = (const float*)d_in[8];
  const float* et           = (const float*)d_in[9];
  float* out  = (float*)d_out;
  float* emis = (float*)d_ws;                      // 16384*3 floats = 192 KB

  init_emis_kernel<<<(MSZ * LSZ + 255) / 256, 256, 0, stream>>>(emis, clsb);
  fused_mlp_kernel<<<dim3(MSZ / MB, 2), TPB, 0, stream>>>(seq, fcw, fcb, clsw, emis);
  crf_nll_kernel<<<1, 32, 0, stream>>>(emis, labels, mask, st, tr, et, out);
}